// SpatialTransformer_2DTo2D_real_64106681860175
// MI455X (gfx1250) — compile-verified
//
#include <hip/hip_runtime.h>
#include <cstdint>

#define FH   144
#define FW   192
#define CH   256
#define GPH  177
#define GPW  152
#define NV   (GPH * GPW)        // 26904
#define NB   4
#define VTOT (NB * NV)          // 107616
#define WPB  4                  // waves per block (wave32)
#define BLOCK (WPB * 32)
#define NBLOCKS 2048

typedef float v4f __attribute__((ext_vector_type(4)));

struct TapInfo {
    const float* g[4];   // per-tap global base (channel 0)
    float        w[4];   // bilinear weights (reference order: a,b,c,d)
};

__device__ __forceinline__ void compute_taps(const float* __restrict__ feats,
                                             const float* __restrict__ im_x,
                                             const float* __restrict__ im_y,
                                             int v, TapInfo& ti) {
    int b = v / NV;
    float x = im_x[v];
    float y = im_y[v];
    x = fminf(fmaxf(x, 0.0f), (float)(FW - 1));
    y = fminf(fmaxf(y, 0.0f), (float)(FH - 1));
    float x0f = floorf(x), y0f = floorf(y);
    int x0 = (int)x0f, y0 = (int)y0f;
    int x1 = min(x0 + 1, FW - 1);
    int y1 = min(y0 + 1, FH - 1);
    float x1f = (float)x1, y1f = (float)y1;
    ti.w[0] = (x1f - x) * (y1f - y);   // Ia @ (y0,x0)
    ti.w[1] = (x1f - x) * (y - y0f);   // Ib @ (y1,x0)
    ti.w[2] = (x - x0f) * (y1f - y);   // Ic @ (y0,x1)
    ti.w[3] = (x - x0f) * (y - y0f);   // Id @ (y1,x1)
    const float* base = feats + (size_t)b * (size_t)FH * FW * CH;
    ti.g[0] = base + ((size_t)y0 * FW + x0) * CH;
    ti.g[1] = base + ((size_t)y1 * FW + x0) * CH;
    ti.g[2] = base + ((size_t)y0 * FW + x1) * CH;
    ti.g[3] = base + ((size_t)y1 * FW + x1) * CH;
}

// Async-gather 4 tap rows (4 x 1KB) into this wave's LDS buffer.
// Lane l moves bytes [16l,16l+16) of each 512B half  ->  8 async b128 ops.
__device__ __forceinline__ void issue_async(const TapInfo& ti, uint32_t ldsBase, int lane) {
#pragma unroll
    for (int t = 0; t < 4; ++t) {
        uint64_t ga0 = (uint64_t)(uintptr_t)(ti.g[t] + lane * 4);   // ch  4l..4l+3
        uint64_t ga1 = ga0 + 512;                                   // ch 128+4l..
        uint32_t la0 = ldsBase + (uint32_t)(t * 1024 + lane * 16);
        uint32_t la1 = la0 + 512;
        asm volatile("global_load_async_to_lds_b128 %0, %1, off"
                     :: "v"(la0), "v"(ga0) : "memory");
        asm volatile("global_load_async_to_lds_b128 %0, %1, off"
                     :: "v"(la1), "v"(ga1) : "memory");
    }
}

__global__ __launch_bounds__(BLOCK)
void bilin_async_kernel(const float* __restrict__ feats,
                        const float* __restrict__ im_x,
                        const float* __restrict__ im_y,
                        const float* __restrict__ mask,
                        float* __restrict__ out) {
    // per-wave double buffer: 2 x (4 taps x 256 ch) fp32 = 8KB/wave, 32KB/block
    __shared__ __attribute__((aligned(16))) float sm[WPB][2][4 * CH];

    const int lane = (int)(threadIdx.x & 31);
    const int wave = (int)(threadIdx.x >> 5);
    const int stride = (int)gridDim.x * WPB;

    int v = (int)blockIdx.x * WPB + wave;
    bool active = v < VTOT;

    TapInfo cur, nxt;
    int bufIdx = 0;

    if (active) {
        compute_taps(feats, im_x, im_y, v, cur);
        issue_async(cur, (uint32_t)(uintptr_t)&sm[wave][0][0], lane);  // 8 in flight
    }

    while (active) {
        const int  vn      = v + stride;
        const bool hasNext = vn < VTOT;

        if (hasNext) {
            compute_taps(feats, im_x, im_y, vn, nxt);
            // make sure our previous-iteration LDS reads drained before the
            // async engine may overwrite that buffer
            asm volatile("s_wait_dscnt 0" ::: "memory");
            issue_async(nxt, (uint32_t)(uintptr_t)&sm[wave][bufIdx ^ 1][0], lane);
            // 16 outstanding; in-order completion => <=8 means vertex v's 8 done
            asm volatile("s_wait_asynccnt 8" ::: "memory");
        } else {
            asm volatile("s_wait_asynccnt 0" ::: "memory");
        }

        const float m = mask[v];
        const float* buf = &sm[wave][bufIdx][0];

        v4f acc0 = (v4f)0.0f, acc1 = (v4f)0.0f;
#pragma unroll
        for (int t = 0; t < 4; ++t) {
            v4f a = *(const v4f*)(buf + t * CH + lane * 4);          // ds_load_b128
            v4f b = *(const v4f*)(buf + t * CH + 128 + lane * 4);
            acc0 += cur.w[t] * a;
            acc1 += cur.w[t] * b;
        }
        acc0 *= m;
        acc1 *= m;

        float* outp = out + (size_t)v * CH;
        __builtin_nontemporal_store(acc0, (v4f*)(outp + lane * 4));        // NT: keep L2 for feats
        __builtin_nontemporal_store(acc1, (v4f*)(outp + 128 + lane * 4));

        cur = nxt;
        v = vn;
        bufIdx ^= 1;
        active = hasNext;
    }
}

extern "C" void kernel_launch(void* const* d_in, const int* in_sizes, int n_in,
                              void* d_out, int out_size, void* d_ws, size_t ws_size,
                              hipStream_t stream) {
    (void)in_sizes; (void)n_in; (void)out_size; (void)d_ws; (void)ws_size;
    const float* feats = (const float*)d_in[0];
    const float* im_x  = (const float*)d_in[1];
    const float* im_y  = (const float*)d_in[2];
    const float* mask  = (const float*)d_in[3];
    float* out = (float*)d_out;

    hipLaunchKernelGGL(bilin_async_kernel, dim3(NBLOCKS), dim3(BLOCK), 0, stream,
                       feats, im_x, im_y, mask, out);
}